// ConceptNet_New_47313359733365
// MI455X (gfx1250) — compile-verified
//
#include <hip/hip_runtime.h>
#include <hip/hip_bf16.h>
#include <math.h>

// ---------------------------------------------------------------------------
// ConceptNet pipeline for MI455X (gfx1250, wave32).
// Heavy passes use V_WMMA_F32_16X16X4_F32 (exact f32 math; the workload is
// HBM-bandwidth bound at ~205MB streamed, so f32 WMMA is the right precision).
// ---------------------------------------------------------------------------

typedef float v2f __attribute__((ext_vector_type(2)));
typedef float v8f __attribute__((ext_vector_type(8)));

#define D_DIM   256
#define NC      32
#define NCLS    10
#define KNN     10
#define FLT_BIG 3.402823466e38f

// ---------------------------------------------------------------------------
// gram = C^T C  [32x32], also c2[j] = gram[j][j]
// C is [256,32] row-major (d-major).
// ---------------------------------------------------------------------------
__global__ void gram_kernel(const float* __restrict__ C,
                            float* __restrict__ gram,
                            float* __restrict__ c2) {
    for (int idx = threadIdx.x; idx < NC * NC; idx += blockDim.x) {
        int i = idx >> 5, j = idx & 31;
        float s = 0.f;
        for (int d = 0; d < D_DIM; ++d)
            s = fmaf(C[d * NC + i], C[d * NC + j], s);
        gram[idx] = s;
        if (i == j) c2[i] = s;
    }
}

// ---------------------------------------------------------------------------
// ginv = gram^-1 via Gauss-Jordan (gram is SPD for random concepts, no pivot).
// Single block, 256 threads, augmented [A | I] in LDS.
// ---------------------------------------------------------------------------
__global__ void inv_kernel(const float* __restrict__ gram,
                           float* __restrict__ ginv) {
    __shared__ float M[NC][2 * NC];
    __shared__ float fac[NC];
    __shared__ float pivinv;
    const int tid = threadIdx.x;

    for (int idx = tid; idx < NC * 2 * NC; idx += blockDim.x) {
        int r = idx >> 6, c = idx & 63;
        M[r][c] = (c < NC) ? gram[r * NC + c] : ((c - NC == r) ? 1.f : 0.f);
    }
    __syncthreads();

    for (int p = 0; p < NC; ++p) {
        if (tid == 0) pivinv = 1.f / M[p][p];
        __syncthreads();
        for (int c = tid; c < 2 * NC; c += blockDim.x) M[p][c] *= pivinv;
        __syncthreads();
        for (int r = tid; r < NC; r += blockDim.x) fac[r] = M[r][p];
        __syncthreads();
        for (int idx = tid; idx < NC * 2 * NC; idx += blockDim.x) {
            int r = idx >> 6, c = idx & 63;
            if (r != p) M[r][c] = fmaf(-fac[r], M[p][c], M[r][c]);
        }
        __syncthreads();
    }
    for (int idx = tid; idx < NC * NC; idx += blockDim.x)
        ginv[idx] = M[idx >> 5][(idx & 31) + NC];
}

// ---------------------------------------------------------------------------
// out[M,32] = A[M,K] @ B[K,32]   (row-major, K multiple of 4, M multiple of 64)
// One wave per 16x32 output block: two v_wmma_f32_16x16x4_f32 accumulators,
// A fragment shared across both N-tiles. 4 waves (128 thr) per block.
// Fragment layout per ISA 7.12.2 (32-bit A 16x4 / B 4x16):
//   lanes 0-15 carry K={k,k+1}, lanes 16-31 carry K={k+2,k+3}.
// ---------------------------------------------------------------------------
__global__ void wmma_gemm32(const float* __restrict__ A,
                            const float* __restrict__ B,
                            float* __restrict__ out, int K) {
    const int lane = threadIdx.x & 31;
    const int wave = threadIdx.x >> 5;
    const int lo = lane & 15, hi = lane >> 4;
    const long tileM = (long)blockIdx.x * 4 + wave;
    const float* arow = A + (tileM * 16 + lo) * (long)K;

    v8f acc0 = {0.f, 0.f, 0.f, 0.f, 0.f, 0.f, 0.f, 0.f};
    v8f acc1 = acc0;
    for (int k = 0; k < K; k += 4) {
        const int ka = k + 2 * hi;
        v2f a = *(const v2f*)(arow + ka);
        v2f b0, b1;
        b0.x = B[ka * NC + lo];        b0.y = B[(ka + 1) * NC + lo];
        b1.x = B[ka * NC + 16 + lo];   b1.y = B[(ka + 1) * NC + 16 + lo];
        acc0 = __builtin_amdgcn_wmma_f32_16x16x4_f32(false, a, false, b0,
                                                     (short)0, acc0, false, false);
        acc1 = __builtin_amdgcn_wmma_f32_16x16x4_f32(false, a, false, b1,
                                                     (short)0, acc1, false, false);
    }
    const long mbase = tileM * 16;
    for (int v = 0; v < 8; ++v) {
        long m = mbase + v + 8 * hi;              // D layout: lane holds N=lo(+16)
        out[m * NC + lo]      = acc0[v];
        out[m * NC + 16 + lo] = acc1[v];
    }
}

// ---------------------------------------------------------------------------
// dist[n,j] = sqrt(max(|E_n|^2 + c2[j] - 2*E_n.C_j, 0))
// Same WMMA core as above with K=256; |E_n|^2 accumulated from the A fragments
// (lane pair combined via shfl_xor 16), so E is streamed exactly once.
// ---------------------------------------------------------------------------
__global__ void dist_wmma(const float* __restrict__ E,
                          const float* __restrict__ C,
                          const float* __restrict__ c2,
                          float* __restrict__ dist) {
    __shared__ float x2s[4][16];
    const int lane = threadIdx.x & 31;
    const int wave = threadIdx.x >> 5;
    const int lo = lane & 15, hi = lane >> 4;
    const long tileM = (long)blockIdx.x * 4 + wave;
    const float* arow = E + (tileM * 16 + lo) * (long)D_DIM;

    v8f acc0 = {0.f, 0.f, 0.f, 0.f, 0.f, 0.f, 0.f, 0.f};
    v8f acc1 = acc0;
    float sq = 0.f;
    for (int k = 0; k < D_DIM; k += 4) {
        const int ka = k + 2 * hi;
        v2f a = *(const v2f*)(arow + ka);
        sq = fmaf(a.x, a.x, sq);
        sq = fmaf(a.y, a.y, sq);
        v2f b0, b1;
        b0.x = C[ka * NC + lo];        b0.y = C[(ka + 1) * NC + lo];
        b1.x = C[ka * NC + 16 + lo];   b1.y = C[(ka + 1) * NC + 16 + lo];
        acc0 = __builtin_amdgcn_wmma_f32_16x16x4_f32(false, a, false, b0,
                                                     (short)0, acc0, false, false);
        acc1 = __builtin_amdgcn_wmma_f32_16x16x4_f32(false, a, false, b1,
                                                     (short)0, acc1, false, false);
    }
    // lanes L and L^16 each hold half of row (lo)'s sum of squares
    float x2 = sq + __shfl_xor(sq, 16, 32);
    if (hi == 0) x2s[wave][lo] = x2;
    __syncthreads();

    const float cj0 = c2[lo], cj1 = c2[16 + lo];
    const long mbase = tileM * 16;
    for (int v = 0; v < 8; ++v) {
        float xr = x2s[wave][v + 8 * hi];
        long  m  = mbase + v + 8 * hi;
        float d0 = fmaf(-2.f, acc0[v], xr + cj0);
        float d1 = fmaf(-2.f, acc1[v], xr + cj1);
        dist[m * NC + lo]      = sqrtf(fmaxf(d0, 0.f));
        dist[m * NC + 16 + lo] = sqrtf(fmaxf(d1, 0.f));
    }
}

// ---------------------------------------------------------------------------
// M1[32,10] = C^T W
// ---------------------------------------------------------------------------
__global__ void ctw_kernel(const float* __restrict__ C,
                           const float* __restrict__ W,
                           float* __restrict__ M1) {
    for (int idx = threadIdx.x; idx < NC * NCLS; idx += blockDim.x) {
        int i = idx / NCLS, c = idx % NCLS;
        float s = 0.f;
        for (int d = 0; d < D_DIM; ++d)
            s = fmaf(C[d * NC + i], W[d * NCLS + c], s);
        M1[idx] = s;
    }
}

// ---------------------------------------------------------------------------
// out[M,NCLS] = A[M,K] @ W[K,NCLS] + b   (classifier heads; tiny)
// ---------------------------------------------------------------------------
__global__ void linear_bias(const float* __restrict__ A,
                            const float* __restrict__ W,
                            const float* __restrict__ b,
                            float* __restrict__ out, int M, int K) {
    int idx = blockIdx.x * blockDim.x + threadIdx.x;
    if (idx >= M * NCLS) return;
    int m = idx / NCLS, c = idx % NCLS;
    const float* a = A + (long)m * K;
    float s = b[c];
    for (int k = 0; k < K; ++k) s = fmaf(a[k], W[k * NCLS + c], s);
    out[idx] = s;
}

// ---------------------------------------------------------------------------
// One block per concept j: top-10 smallest dist[:,j] (per-thread sorted lists
// + 10 argmin rounds in LDS), then dots[j] = (1/k) sum_t E[idx_t] . C[:,j].
// blockDim must be 256 (== D_DIM for the dot phase).
// ---------------------------------------------------------------------------
__global__ void topk_dots(const float* __restrict__ dist,
                          const float* __restrict__ E,
                          const float* __restrict__ C,
                          float* __restrict__ dots, int N) {
    __shared__ float sval[256 * KNN];
    __shared__ int   sidx[256 * KNN];
    __shared__ float rval[256];
    __shared__ int   ridx[256];
    __shared__ int   knn[KNN];

    const int tid = threadIdx.x;
    const int j = blockIdx.x;

    float best[KNN];
    int   bidx[KNN];
    for (int t = 0; t < KNN; ++t) { best[t] = FLT_BIG; bidx[t] = 0; }

    for (long n = tid; n < N; n += blockDim.x) {
        float d = dist[n * NC + j];
        if (d < best[KNN - 1]) {
            int p = KNN - 1;
            while (p > 0 && best[p - 1] > d) {
                best[p] = best[p - 1]; bidx[p] = bidx[p - 1]; --p;
            }
            best[p] = d; bidx[p] = (int)n;
        }
    }
    for (int t = 0; t < KNN; ++t) {
        sval[tid * KNN + t] = best[t];
        sidx[tid * KNN + t] = bidx[t];
    }
    __syncthreads();

    for (int t = 0; t < KNN; ++t) {
        float mv = FLT_BIG; int mi = 0;
        for (int i = tid; i < 256 * KNN; i += blockDim.x)
            if (sval[i] < mv) { mv = sval[i]; mi = i; }
        rval[tid] = mv; ridx[tid] = mi;
        __syncthreads();
        for (int s = 128; s > 0; s >>= 1) {
            if (tid < s && rval[tid + s] < rval[tid]) {
                rval[tid] = rval[tid + s]; ridx[tid] = ridx[tid + s];
            }
            __syncthreads();
        }
        if (tid == 0) {
            knn[t] = sidx[ridx[0]];
            sval[ridx[0]] = FLT_BIG;        // remove winner for next round
        }
        __syncthreads();
    }

    // dots[j] = sum_d C[d,j] * (sum_t E[knn_t, d]) / k   (tid == d)
    float s = 0.f;
    for (int t = 0; t < KNN; ++t)
        s += E[(long)knn[t] * D_DIM + tid];
    rval[tid] = s * C[tid * NC + j];
    __syncthreads();
    for (int st = 128; st > 0; st >>= 1) {
        if (tid < st) rval[tid] += rval[tid + st];
        __syncthreads();
    }
    if (tid == 0) dots[j] = rval[0] / (float)KNN;
}

// ---------------------------------------------------------------------------
// Final scalars: L1 = mean(dots); L2 = sum_offdiag(gram)/1024; nm = trace/1024
// ---------------------------------------------------------------------------
__global__ void finalize_kernel(const float* __restrict__ gram,
                                const float* __restrict__ dots,
                                float* __restrict__ out3) {
    __shared__ float sOff[NC], sDiag[NC], sDots[NC];
    int t = threadIdx.x;
    if (t < NC) {
        float off = 0.f, dg = 0.f;
        for (int j = 0; j < NC; ++j) {
            float g = gram[t * NC + j];
            if (j == t) dg = g; else off += g;
        }
        sOff[t] = off; sDiag[t] = dg; sDots[t] = dots[t];
    }
    __syncthreads();
    if (t == 0) {
        float o = 0.f, d = 0.f, ds = 0.f;
        for (int i = 0; i < NC; ++i) { o += sOff[i]; d += sDiag[i]; ds += sDots[i]; }
        out3[0] = ds / (float)NC;                 // L_sparse_1
        out3[1] = o / (float)(NC * NC);           // L_sparse_2
        out3[2] = d / (float)(NC * NC);           // norm_metrics
    }
}

// ---------------------------------------------------------------------------
extern "C" void kernel_launch(void* const* d_in, const int* in_sizes, int n_in,
                              void* d_out, int out_size, void* d_ws, size_t ws_size,
                              hipStream_t stream) {
    const float* X = (const float*)d_in[0];   // [4096, 256]
    const float* E = (const float*)d_in[1];   // [N, 256]
    const float* C = (const float*)d_in[2];   // [256, 32]
    const float* W = (const float*)d_in[3];   // [256, 10]
    const float* b = (const float*)d_in[4];   // [10]

    const int BS = in_sizes[0] / D_DIM;       // 4096
    const int N  = in_sizes[1] / D_DIM;       // 200000

    float* out = (float*)d_out;
    float* orig_pred = out;                   // [BS, 10]
    float* y_pred    = out + (long)BS * NCLS; // [BS, 10]
    float* out3      = out + 2L * BS * NCLS;  // 3 scalars

    // workspace layout (floats)
    float* ws   = (float*)d_ws;
    float* gram = ws;                          // 1024
    float* ginv = ws + 1024;                   // 1024
    float* c2   = ws + 2048;                   // 32
    float* dots = ws + 2080;                   // 32
    float* M1   = ws + 2112;                   // 320
    float* T1   = ws + 4096;                   // BS*32
    float* T2   = T1 + (long)BS * NC;          // BS*32
    float* dist = T2 + (long)BS * NC;          // N*32

    // 1) gram + c2, then invert
    gram_kernel<<<1, 256, 0, stream>>>(C, gram, c2);
    inv_kernel<<<1, 256, 0, stream>>>(gram, ginv);

    // 2) T1 = X @ C ; T2 = T1 @ ginv   (WMMA f32)
    wmma_gemm32<<<BS / 64, 128, 0, stream>>>(X, C, T1, D_DIM);
    wmma_gemm32<<<BS / 64, 128, 0, stream>>>(T1, ginv, T2, NC);

    // 3) classifier heads
    ctw_kernel<<<1, 256, 0, stream>>>(C, W, M1);
    linear_bias<<<(BS * NCLS + 255) / 256, 256, 0, stream>>>(X, W, b, orig_pred, BS, D_DIM);
    linear_bias<<<(BS * NCLS + 255) / 256, 256, 0, stream>>>(T2, M1, b, y_pred, BS, NC);

    // 4) fused distance pass over E (single stream of 205MB, WMMA f32 core)
    dist_wmma<<<N / 64, 128, 0, stream>>>(E, C, c2, dist);

    // 5) per-concept top-k + knn dots, then final scalar reductions
    topk_dots<<<NC, 256, 0, stream>>>(dist, E, C, dots, N);
    finalize_kernel<<<1, 32, 0, stream>>>(gram, dots, out3);
}